// TransE_26774826123521
// MI455X (gfx1250) — compile-verified
//
#include <hip/hip_runtime.h>

// TransE kvsall scoring on gfx1250 (MI455X, wave32, WMMA).
// score[b,e] = || (E[h_b] + R[r_b]) - E[e] ||_2
// Decomposed as ||hr||^2 + ||e||^2 - 2*(hr . e); the cross term is a
// 32x100000x128 f32 GEMM done with V_WMMA_F32_16X16X4_F32.

typedef __attribute__((ext_vector_type(2))) float v2f;
typedef __attribute__((ext_vector_type(8))) float v8f;

#define ENT_TOT 100000
#define DIM 128
#define BATCH 32
#define NTILES (ENT_TOT / 16) /* 6250, exact */
#define WPB 8                 /* waves per block */

// ---------------------------------------------------------------------------
// Kernel 1: hr[b][d] = ent[h[b]][d] + rel[r[b]][d];  hrn[b] = ||hr[b]||^2
// grid = BATCH blocks, block = DIM threads. Fully rewrites its workspace
// region every call (workspace is poisoned by the harness).
// ---------------------------------------------------------------------------
__global__ __launch_bounds__(DIM) void transe_prep(
    const int* __restrict__ h, const int* __restrict__ r,
    const float* __restrict__ ent, const float* __restrict__ rel,
    float* __restrict__ hr, float* __restrict__ hrn)
{
    __shared__ float red[DIM];
    const int b = blockIdx.x;
    const int d = threadIdx.x;
    const float v = ent[(size_t)h[b] * DIM + d] + rel[(size_t)r[b] * DIM + d];
    hr[b * DIM + d] = v;
    red[d] = v * v;
    __syncthreads();
#pragma unroll
    for (int s = DIM / 2; s > 0; s >>= 1) {
        if (d < s) red[d] += red[d + s];
        __syncthreads();
    }
    if (d == 0) hrn[b] = red[0];
}

// ---------------------------------------------------------------------------
// Kernel 2: each wave computes a 32x16 output tile (all 32 batch rows x 16
// entities) with two f32 WMMA accumulators over K=128.
//
// VGPR layouts (CDNA5 ISA 7.12.2), both reduce to the same per-lane rule:
//   A 16x4 f32 : lane l, vgpr j -> m = l&15,  k = k0 + j + 2*(l>>4)
//   B 4x16 f32 : lane l, vgpr j -> n = l&15,  k = k0 + j + 2*(l>>4)
//   C 16x16 f32: lane l, vgpr v -> n = l&15,  m = v + 8*(l>>4)
// so A and B fragments are single 8-byte (v2f) loads per k-step.
// ---------------------------------------------------------------------------
__global__ __launch_bounds__(32 * WPB) void transe_score(
    const float* __restrict__ ent, const float* __restrict__ hr,
    const float* __restrict__ hrn, float* __restrict__ out)
{
    __shared__ float s_hr[BATCH * DIM];
    __shared__ float s_hrn[BATCH];

    for (int i = threadIdx.x; i < BATCH * DIM; i += 32 * WPB) s_hr[i] = hr[i];
    if (threadIdx.x < BATCH) s_hrn[threadIdx.x] = hrn[threadIdx.x];
    __syncthreads();

    const int lane = threadIdx.x & 31;
    const int wave = threadIdx.x >> 5;
    const int tile = blockIdx.x * WPB + wave;
    if (tile >= NTILES) return; // wave-uniform: EXEC stays all-1s for WMMA

    const int n  = lane & 15; // entity column within tile (== batch row l&15)
    const int hi = lane >> 4; // half-wave select

    const float* erow = ent + (size_t)(tile * 16 + n) * DIM;

    v8f c0 = {}; // rows m = 0..15
    v8f c1 = {}; // rows m = 16..31
    float en_part = 0.0f;

#pragma unroll
    for (int k0 = 0; k0 < DIM; k0 += 4) {
        const int k = k0 + 2 * hi;
        const v2f bf = *(const v2f*)(erow + k);              // global_load_b64
        en_part += bf.x * bf.x + bf.y * bf.y;                // entity norm piece
        const v2f a0 = *(const v2f*)(&s_hr[n * DIM + k]);        // ds_load_b64
        const v2f a1 = *(const v2f*)(&s_hr[(16 + n) * DIM + k]); // ds_load_b64
        c0 = __builtin_amdgcn_wmma_f32_16x16x4_f32(
            false, a0, false, bf, (short)0, c0, false, false);
        c1 = __builtin_amdgcn_wmma_f32_16x16x4_f32(
            false, a1, false, bf, (short)0, c1, false, false);
    }

    // lanes l and l+16 together hold the full ||E[n]||^2 for n = l&15
    const float en = en_part + __shfl_xor(en_part, 16, 32);

    float* ocol = out + (size_t)tile * 16 + n;
#pragma unroll
    for (int v = 0; v < 8; ++v) {
        const int m = v + 8 * hi;
        const float s0 = s_hrn[m] + en - 2.0f * c0[v];
        ocol[(size_t)m * ENT_TOT] = sqrtf(fmaxf(s0, 0.0f));
        const float s1 = s_hrn[m + 16] + en - 2.0f * c1[v];
        ocol[(size_t)(m + 16) * ENT_TOT] = sqrtf(fmaxf(s1, 0.0f));
    }
}

// ---------------------------------------------------------------------------
// Host launcher. Inputs (setup_inputs order): h, r, t, ent_emb, rel_emb,
// batch_size. Workspace layout: hr (32*128 f32) | hrn (32 f32).
// ---------------------------------------------------------------------------
extern "C" void kernel_launch(void* const* d_in, const int* in_sizes, int n_in,
                              void* d_out, int out_size, void* d_ws, size_t ws_size,
                              hipStream_t stream) {
    (void)in_sizes; (void)n_in; (void)out_size; (void)ws_size;
    const int*   h   = (const int*)d_in[0];
    const int*   r   = (const int*)d_in[1];
    const float* ent = (const float*)d_in[3];
    const float* rel = (const float*)d_in[4];
    float* out = (float*)d_out;

    float* hr  = (float*)d_ws;          // 32*128 floats
    float* hrn = hr + BATCH * DIM;      // 32 floats

    transe_prep<<<BATCH, DIM, 0, stream>>>(h, r, ent, rel, hr, hrn);

    const int nblocks = (NTILES + WPB - 1) / WPB; // 782
    transe_score<<<nblocks, 32 * WPB, 0, stream>>>(ent, hr, hrn, out);
}